// ToolDetectionEnsemble_39333310497281
// MI455X (gfx1250) — compile-verified
//
#include <hip/hip_runtime.h>
#include <stdint.h>

// ---------------- problem constants ----------------
#define MDL      3
#define BIMG     16
#define NBOX     512
#define KTOT     1536          // MDL*NBOX boxes per image
#define KPAD     2048          // bitonic pad
#define NWORD    48            // KTOT/32 bitmask words per full row
#define NTHREADS 1024          // 32 waves of 32
#define IOU_THRF 0.5f

typedef unsigned int u32;
typedef unsigned long long u64;
typedef __attribute__((ext_vector_type(4))) u32 v4u;
typedef __attribute__((ext_vector_type(8))) int v8i;
typedef __attribute__((ext_vector_type(4))) int v4i;

// ---------------- LDS layout (byte offsets) ----------------
// Region A (persistent after gather):
#define OFF_BOX    0u                       // 1536 * float4      = 24576
#define OFF_SCORE  24576u                   // 1536 * f32         = 6144
#define OFF_LABEL  30720u                   // 1536 * i32         = 6144
#define OFF_KEEP   36864u                   // 1536 * u32         = 6144
// Region B: triangular suppression bitmask, sum_i (48 - i/32) words = 37632 u32
#define OFF_MASK   43008u                   // 37632 * u32        = 150528
// Region B is reused for raw staging + sort keys BEFORE the mask is built:
#define RAW_BOX    (OFF_MASK + 0u)          // 24576
#define RAW_SCORE  (OFF_MASK + 24576u)      // 6144
#define RAW_LABEL  (OFF_MASK + 30720u)      // 6144
#define OFF_KEY    (OFF_MASK + 36864u)      // 2048 * u64 = 16384 (<= 150528, fits)
#define LDS_BYTES  (OFF_MASK + 150528u)     // 193536 bytes < 320 KB/WGP

#if defined(__AMDGCN__) && __has_builtin(__builtin_amdgcn_tensor_load_to_lds)
#define USE_TDM 1
#else
#define USE_TDM 0
#endif

#if defined(__AMDGCN__) && __has_builtin(__builtin_amdgcn_readlane)
#define HAS_READLANE 1
#else
#define HAS_READLANE 0
#endif

#if USE_TDM
// Issue one TDM descriptor: 2-D tile of 4-byte elements, `nlines` lines of
// `dim0_elems` contiguous elements, line stride `stride_elems` (all in elems).
// D# bit layout per CDNA5 ISA ch.8 (group0 128b, group1 256b, groups2/3 zero).
// 6-arg builtin on this toolchain:
//   (v4u group0, v8i group1, v4i group2, v4i group3, v8i group4(unused), i32 cpol)
__device__ __attribute__((always_inline)) void tdm_load_2d(
    u32 lds_off, const void* gptr, u32 dim0_elems, u32 nlines, u32 stride_elems) {
  u64 ga = (u64)(uintptr_t)gptr;
  v4u g0;
  g0.x = 1u;                                            // count=1 valid descriptor
  g0.y = lds_off;                                       // lds_addr (bytes)
  g0.z = (u32)ga;                                       // global_addr[31:0]
  g0.w = (u32)((ga >> 32) & 0x01FFFFFFu) | (2u << 30);  // global_addr[56:32] | type=2
  v8i g1;
  g1[0] = (int)(2u << 16);                                          // data_size=4B
  g1[1] = (int)((dim0_elems & 0xFFFFu) << 16);                      // tensor_dim0[15:0]
  g1[2] = (int)((dim0_elems >> 16) | ((nlines & 0xFFFFu) << 16));   // dim0 hi | tensor_dim1 lo
  g1[3] = (int)((nlines >> 16) | ((dim0_elems & 0xFFFFu) << 16));   // dim1 hi | tile_dim0
  g1[4] = (int)(nlines & 0xFFFFu);                                  // tile_dim1 | tile_dim2=0
  g1[5] = (int)stride_elems;                                        // tensor_dim0_stride[31:0]
  g1[6] = 0;                                                        // stride hi | dim1_stride lo
  g1[7] = 0;
  v4i z4; z4[0] = 0; z4[1] = 0; z4[2] = 0; z4[3] = 0;
  v8i z8; z8[0] = 0; z8[1] = 0; z8[2] = 0; z8[3] = 0;
  z8[4] = 0; z8[5] = 0; z8[6] = 0; z8[7] = 0;
  __builtin_amdgcn_tensor_load_to_lds(g0, g1, z4, z4, z8, 0);
}
#endif

__device__ __attribute__((always_inline)) int mask_row_base(int i) {
  int w0 = i >> 5;
  // 32 rows per word-block; block w stores (48-w) words per row.
  return ((48 * w0 - ((w0 * (w0 - 1)) >> 1)) << 5) + (i & 31) * (48 - w0);
}

__global__ __launch_bounds__(NTHREADS, 1)
void ensemble_nms_kernel(const float* __restrict__ boxes,
                         const float* __restrict__ scores,
                         const int* __restrict__ labels,
                         const float* __restrict__ weights,
                         float* __restrict__ out) {
  __shared__ __align__(16) unsigned char smem[LDS_BYTES];
  const int tid = (int)threadIdx.x;
  const int img = (int)blockIdx.x;

  float4* sBox   = (float4*)(smem + OFF_BOX);
  float*  sScore = (float*) (smem + OFF_SCORE);
  int*    sLabel = (int*)   (smem + OFF_LABEL);
  u32*    sKeep  = (u32*)   (smem + OFF_KEEP);
  u32*    sMask  = (u32*)   (smem + OFF_MASK);
  float4* rBox   = (float4*)(smem + RAW_BOX);
  float*  rScore = (float*) (smem + RAW_SCORE);
  int*    rLabel = (int*)   (smem + RAW_LABEL);
  u64*    sKey   = (u64*)   (smem + OFF_KEY);

  // ---- Stage this image's boxes/scores/labels into LDS via the TDM ----
#if USE_TDM
  if (tid == 0) {
    // boxes (M,B,N,4): line m at elem offset m*B*N*4, image offset img*N*4
    tdm_load_2d((u32)(uintptr_t)(smem + RAW_BOX),
                boxes + (size_t)img * NBOX * 4, NBOX * 4, MDL,
                (u32)(BIMG * NBOX * 4));
    tdm_load_2d((u32)(uintptr_t)(smem + RAW_SCORE),
                scores + (size_t)img * NBOX, NBOX, MDL, (u32)(BIMG * NBOX));
    tdm_load_2d((u32)(uintptr_t)(smem + RAW_LABEL),
                labels + (size_t)img * NBOX, NBOX, MDL, (u32)(BIMG * NBOX));
    __builtin_amdgcn_s_wait_tensorcnt((short)0);
  }
#else
  for (int i = tid; i < KTOT; i += NTHREADS) {
    int m = i >> 9, n = i & (NBOX - 1);
    size_t g = ((size_t)m * BIMG + img) * NBOX + n;
    rBox[i]   = ((const float4*)boxes)[g];
    rScore[i] = scores[g];
    rLabel[i] = labels[g];
  }
#endif
  __syncthreads();

  // ---- Build sort keys: score desc (non-negative floats), tie -> idx asc ----
  float wv0 = weights[0], wv1 = weights[1], wv2 = weights[2];
  for (int i = tid; i < KPAD; i += NTHREADS) {
    u64 key;
    if (i < KTOT) {
      int m = i >> 9;
      float w = (m == 0) ? wv0 : (m == 1 ? wv1 : wv2);
      u32 sb = __float_as_uint(rScore[i] * w);
      key = ((u64)(~sb) << 32) | (u32)i;
    } else {
      key = ~0ull;  // pad sorts last
    }
    sKey[i] = key;
  }
  __syncthreads();

  // ---- Bitonic sort of 2048 keys, ascending (== score descending) ----
  for (int k = 2; k <= KPAD; k <<= 1) {
    for (int j = k >> 1; j > 0; j >>= 1) {
      int i0 = (tid & (j - 1)) | ((tid & ~(j - 1)) << 1);
      int i1 = i0 | j;
      u64 a = sKey[i0], c = sKey[i1];
      bool up = (i0 & k) == 0;
      if ((a > c) == up) { sKey[i0] = c; sKey[i1] = a; }
      __syncthreads();
    }
  }

  // ---- Gather into sorted order (region A); scores recovered from key bits ----
  for (int i = tid; i < KTOT; i += NTHREADS) {
    u64 key = sKey[i];
    int idx = (int)(u32)key;
    sScore[i] = __uint_as_float(~(u32)(key >> 32));
    sLabel[i] = rLabel[idx];
    sBox[i]   = rBox[idx];
  }
  __syncthreads();  // raw staging + keys dead from here; region B becomes mask

  // ---- Triangular suppression bitmask: one wave per row, one word per pass ----
  {
    const int wave = tid >> 5, lane = tid & 31;
    for (int i = wave; i < KTOT; i += 32) {      // 48 rows per wave, balanced load
      float4 bi = sBox[i];
      int   li = sLabel[i];
      float ai = (bi.z - bi.x) * (bi.w - bi.y);
      int   w0 = i >> 5;
      int base = mask_row_base(i);
      for (int cw = w0; cw < NWORD; ++cw) {
        int j = (cw << 5) | lane;
        bool sup = false;
        if (j > i) {
          float4 bj = sBox[j];
          float aj = (bj.z - bj.x) * (bj.w - bj.y);
          float iw = fminf(bi.z, bj.z) - fmaxf(bi.x, bj.x);
          float ih = fminf(bi.w, bj.w) - fmaxf(bi.y, bj.y);
          iw = fmaxf(iw, 0.0f); ih = fmaxf(ih, 0.0f);
          float inter = iw * ih;
          float uni = ai + aj - inter;
          // inter / max(uni,1e-9) > THR  <=>  inter > THR * max(uni,1e-9)
          sup = (inter > IOU_THRF * fmaxf(uni, 1e-9f)) && (sLabel[j] == li);
        }
        u64 bal = __ballot(sup);
        if (lane == 0) sMask[base + (cw - w0)] = (u32)bal;
      }
    }
  }
  __syncthreads();

  // ---- Serial greedy scan (wave 0): remv held in registers, decision bit
  //      broadcast via v_readlane; next row's mask words prefetched from LDS
  //      unconditionally so DS latency stays off the dependence chain. ----
  if (tid < 32) {
    const int lane = tid;
    u32 r0 = 0u, r1 = 0u;                     // remv words: lane, lane+32
    u32 m0 = sMask[lane];                     // row 0: base 0, w0 = 0
    u32 m1 = (lane < NWORD - 32) ? sMask[32 + lane] : 0u;
    for (int i = 0; i < KTOT; ++i) {
      int w = i >> 5;
#if HAS_READLANE
      u32 word = (w < 32) ? (u32)__builtin_amdgcn_readlane((int)r0, w)
                          : (u32)__builtin_amdgcn_readlane((int)r1, w - 32);
#else
      u32 word = (w < 32) ? (u32)__shfl((int)r0, w, 32)
                          : (u32)__shfl((int)r1, w - 32, 32);
#endif
      u32 kept = ((word >> (i & 31)) & 1u) ^ 1u;
      if (lane == 0) sKeep[i] = kept;
      if (kept) { r0 |= m0; r1 |= m1; }       // uniform branch
      int ni = i + 1;
      u32 t0 = 0u, t1 = 0u;
      if (ni < KTOT) {                        // prefetch row ni's mask words
        int w0n = ni >> 5;
        int bn = mask_row_base(ni);
        if (lane >= w0n) t0 = sMask[bn + (lane - w0n)];
        if (lane < NWORD - 32 && (lane + 32) >= w0n)
          t1 = sMask[bn + (lane + 32 - w0n)];
      }
      m0 = t0; m1 = t1;
    }
  }
  __syncthreads();

  // ---- Emit (B, K, 5): [box*keep, score*keep] in sorted order ----
  float* op = out + (size_t)img * (KTOT * 5);
  for (int i = tid; i < KTOT; i += NTHREADS) {
    float kf = sKeep[i] ? 1.0f : 0.0f;
    float4 bx = sBox[i];
    int o = i * 5;
    op[o + 0] = bx.x * kf;
    op[o + 1] = bx.y * kf;
    op[o + 2] = bx.z * kf;
    op[o + 3] = bx.w * kf;
    op[o + 4] = sScore[i] * kf;
  }
}

extern "C" void kernel_launch(void* const* d_in, const int* in_sizes, int n_in,
                              void* d_out, int out_size, void* d_ws, size_t ws_size,
                              hipStream_t stream) {
  (void)in_sizes; (void)n_in; (void)d_ws; (void)ws_size; (void)out_size;
  const float* boxes   = (const float*)d_in[0];
  const float* scores  = (const float*)d_in[1];
  const int*   labels  = (const int*)d_in[2];
  const float* weights = (const float*)d_in[3];
  float* out = (float*)d_out;
  ensemble_nms_kernel<<<dim3(BIMG), dim3(NTHREADS), 0, stream>>>(
      boxes, scores, labels, weights, out);
}